// GATNetwork_80126909874312
// MI455X (gfx1250) — compile-verified
//
#include <hip/hip_runtime.h>
#include <math.h>

#define N_NODES   50000
#define N_EDGES   800000
#define E_TOT     (N_EDGES + N_NODES)   // with self loops
#define D_FEAT    128
#define HEADS     4
#define HIDDEN    32
#define NCLS      24
#define NCLS_PAD  32
#define NEG_SLOPE 0.2f

typedef __attribute__((ext_vector_type(2))) float v2f;
typedef __attribute__((ext_vector_type(8))) float v8f;
typedef int v4i __attribute__((vector_size(16)));

#define __AS(n) __attribute__((address_space(n)))

#if __has_builtin(__builtin_amdgcn_global_load_async_to_lds_b128) && \
    __has_builtin(__builtin_amdgcn_s_wait_asynccnt)
#define USE_ASYNC_LDS 1
#else
#define USE_ASYNC_LDS 0
#endif

// ---------- helpers ----------

__device__ __forceinline__ unsigned enc_f32(float f) {
    unsigned u = __float_as_uint(f);
    return (u & 0x80000000u) ? ~u : (u | 0x80000000u);
}
__device__ __forceinline__ float dec_f32(unsigned k) {
    return (k & 0x80000000u) ? __uint_as_float(k & 0x7FFFFFFFu)
                             : __uint_as_float(~k);
}
__device__ __forceinline__ float lrelu(float v) {
    return v > 0.0f ? v : NEG_SLOPE * v;
}
__device__ __forceinline__ void get_edge(const int* __restrict__ ei, int e,
                                         int& s, int& d) {
    if (e < N_EDGES) { s = ei[e]; d = ei[N_EDGES + e]; }
    else             { s = d = e - N_EDGES; }
}

// ---------- WMMA fp32 GEMM: C[M x COLT*16] = A[M x K] * B[K x COLT*16] ----------
// B (K x COLT*16, contiguous, ldb == COLT*16) is staged once per block into LDS
// via async global->LDS copies; one wave then computes a 16-row strip with
// COLT 16x16 f32 accumulator tiles held in registers.
// LDS row stride = ldb+8 so rows k and k+2 (the two lane-halves of a B fragment)
// land in disjoint bank ranges: 2*(ldb+8) == 16 (mod 64).
template <int COLT>
__global__ void wmma_gemm_f32(const float* __restrict__ A, int lda,
                              const float* __restrict__ B,
                              float* __restrict__ C, int ldc,
                              int M, int K) {
    constexpr int LDB  = COLT * 16;
    constexpr int LDBP = LDB + 8;           // padded LDS row stride
    extern __shared__ float sB[];           // K * LDBP floats

    // ---- stage B into LDS (all threads participate, even in partial blocks) ----
    {
        const int nf4 = (D_FEAT * LDB) / 4; // K == D_FEAT for both layers
        for (int i = threadIdx.x; i < nf4; i += blockDim.x) {
            int row = i / (LDB / 4);
            int c4  = i % (LDB / 4);
            const float* gp = B + (size_t)row * LDB + c4 * 4;
            float*       lp = sB + row * LDBP + c4 * 4;
#if USE_ASYNC_LDS
            __builtin_amdgcn_global_load_async_to_lds_b128(
                (__AS(1) v4i*)(__AS(1) void*)gp,
                (__AS(3) v4i*)(__AS(3) void*)lp,
                0, 0);
#else
            *(float4*)lp = *(const float4*)gp;
#endif
        }
#if USE_ASYNC_LDS
        __builtin_amdgcn_s_wait_asynccnt(0);
#endif
        __syncthreads();
    }

    int wave = (blockIdx.x * blockDim.x + threadIdx.x) >> 5;
    int lane = threadIdx.x & 31;
    int row0 = wave * 16;
    if (row0 >= M) return;

    const int half = lane >> 4;   // 0: lanes 0-15, 1: lanes 16-31
    const int l16  = lane & 15;

    v8f acc[COLT];
    for (int t = 0; t < COLT; ++t)
        for (int r = 0; r < 8; ++r) acc[t][r] = 0.0f;

    for (int k0 = 0; k0 < K; k0 += 4) {
        // A fragment: lane half 0 holds K=k0,k0+1 ; half 1 holds K=k0+2,k0+3
        const float* ap = A + (size_t)(row0 + l16) * lda + k0 + half * 2;
        v2f a; a.x = ap[0]; a.y = ap[1];
        for (int t = 0; t < COLT; ++t) {
            // B fragment from LDS: VGPR0 = row (k0 + 2*half), VGPR1 = next row
            const float* bp = sB + (k0 + half * 2) * LDBP + t * 16 + l16;
            v2f b; b.x = bp[0]; b.y = bp[LDBP];
            acc[t] = __builtin_amdgcn_wmma_f32_16x16x4_f32(
                false, a, false, b, (short)0, acc[t], false, false);
        }
    }

    for (int t = 0; t < COLT; ++t)
        for (int r = 0; r < 8; ++r) {
            int row = row0 + r + half * 8;       // C VGPR r: M=r (lo half), M=r+8 (hi half)
            C[(size_t)row * ldc + t * 16 + l16] = acc[t][r];
        }
}

// ---------- small kernels ----------

__global__ void build_w2p_kernel(const float* __restrict__ W2, float* __restrict__ W2p) {
    int t = blockIdx.x * blockDim.x + threadIdx.x;
    if (t >= D_FEAT * NCLS_PAD) return;
    int r = t / NCLS_PAD, c = t % NCLS_PAD;
    W2p[t] = (c < NCLS) ? W2[r * NCLS + c] : 0.0f;
}

__global__ void al1_kernel(const float* __restrict__ h1,
                           const float* __restrict__ a_src,
                           const float* __restrict__ a_dst,
                           float* __restrict__ al_s, float* __restrict__ al_d) {
    int t = blockIdx.x * blockDim.x + threadIdx.x;
    if (t >= N_NODES * HEADS) return;
    int n = t >> 2, h = t & 3;
    const float* hp = h1 + (size_t)n * D_FEAT + h * HIDDEN;
    const float* as = a_src + h * HIDDEN;
    const float* ad = a_dst + h * HIDDEN;
    float ss = 0.0f, sd = 0.0f;
    for (int c = 0; c < HIDDEN; ++c) { float v = hp[c]; ss += v * as[c]; sd += v * ad[c]; }
    al_s[t] = ss; al_d[t] = sd;
}

__global__ void e1_max_kernel(const int* __restrict__ ei,
                              const float* __restrict__ al_s,
                              const float* __restrict__ al_d,
                              unsigned* __restrict__ emax) {
    int e = blockIdx.x * blockDim.x + threadIdx.x;
    if (e >= E_TOT) return;
    int s, d; get_edge(ei, e, s, d);
    for (int h = 0; h < HEADS; ++h) {
        float v = lrelu(al_s[s * HEADS + h] + al_d[d * HEADS + h]);
        atomicMax(&emax[d * HEADS + h], enc_f32(v));
    }
}

__global__ void e1_sum_kernel(const int* __restrict__ ei,
                              const float* __restrict__ al_s,
                              const float* __restrict__ al_d,
                              const unsigned* __restrict__ emax,
                              float* __restrict__ denom) {
    int e = blockIdx.x * blockDim.x + threadIdx.x;
    if (e >= E_TOT) return;
    int s, d; get_edge(ei, e, s, d);
    for (int h = 0; h < HEADS; ++h) {
        float v = lrelu(al_s[s * HEADS + h] + al_d[d * HEADS + h]);
        float ex = expf(v - dec_f32(emax[d * HEADS + h]));
        atomicAdd(&denom[d * HEADS + h], ex);
    }
}

// 32 lanes per edge; lane handles channel `lane` of each of the 4 heads.
__global__ void e1_agg_kernel(const int* __restrict__ ei,
                              const float* __restrict__ al_s,
                              const float* __restrict__ al_d,
                              const unsigned* __restrict__ emax,
                              const float* __restrict__ denom,
                              const float* __restrict__ h1,
                              float* __restrict__ agg) {
    int t = blockIdx.x * blockDim.x + threadIdx.x;
    int e = t >> 5;
    if (e >= E_TOT) return;
    int lane = t & 31;
    int s, d; get_edge(ei, e, s, d);
    float alpha[HEADS];
    for (int h = 0; h < HEADS; ++h) {
        float v  = lrelu(al_s[s * HEADS + h] + al_d[d * HEADS + h]);
        float ex = expf(v - dec_f32(emax[d * HEADS + h]));
        alpha[h] = ex / (denom[d * HEADS + h] + 1e-16f);
    }
    for (int h = 0; h < HEADS; ++h) {
        float val = alpha[h] * h1[(size_t)s * D_FEAT + h * HIDDEN + lane];
        atomicAdd(&agg[(size_t)d * D_FEAT + h * HIDDEN + lane], val);
    }
}

__global__ void h2in_kernel(float* __restrict__ agg, const float* __restrict__ b1) {
    int t = blockIdx.x * blockDim.x + threadIdx.x;
    if (t >= N_NODES * D_FEAT) return;
    float v = agg[t] + b1[t & (D_FEAT - 1)];
    agg[t] = v > 0.0f ? v : (expf(v) - 1.0f);     // ELU, alpha=1
}

__global__ void al2_kernel(const float* __restrict__ h2,
                           const float* __restrict__ a_src,
                           const float* __restrict__ a_dst,
                           float* __restrict__ al_s, float* __restrict__ al_d) {
    int n = blockIdx.x * blockDim.x + threadIdx.x;
    if (n >= N_NODES) return;
    const float* hp = h2 + (size_t)n * NCLS_PAD;
    float ss = 0.0f, sd = 0.0f;
    for (int c = 0; c < NCLS; ++c) { float v = hp[c]; ss += v * a_src[c]; sd += v * a_dst[c]; }
    al_s[n] = ss; al_d[n] = sd;
}

__global__ void e2_max_kernel(const int* __restrict__ ei,
                              const float* __restrict__ al_s,
                              const float* __restrict__ al_d,
                              unsigned* __restrict__ emax) {
    int e = blockIdx.x * blockDim.x + threadIdx.x;
    if (e >= E_TOT) return;
    int s, d; get_edge(ei, e, s, d);
    atomicMax(&emax[d], enc_f32(lrelu(al_s[s] + al_d[d])));
}

__global__ void e2_sum_kernel(const int* __restrict__ ei,
                              const float* __restrict__ al_s,
                              const float* __restrict__ al_d,
                              const unsigned* __restrict__ emax,
                              float* __restrict__ denom) {
    int e = blockIdx.x * blockDim.x + threadIdx.x;
    if (e >= E_TOT) return;
    int s, d; get_edge(ei, e, s, d);
    float ex = expf(lrelu(al_s[s] + al_d[d]) - dec_f32(emax[d]));
    atomicAdd(&denom[d], ex);
}

__global__ void e2_agg_kernel(const int* __restrict__ ei,
                              const float* __restrict__ al_s,
                              const float* __restrict__ al_d,
                              const unsigned* __restrict__ emax,
                              const float* __restrict__ denom,
                              const float* __restrict__ h2,
                              float* __restrict__ agg) {
    int t = blockIdx.x * blockDim.x + threadIdx.x;
    int e = t >> 5;
    if (e >= E_TOT) return;
    int lane = t & 31;
    int s, d; get_edge(ei, e, s, d);
    float ex = expf(lrelu(al_s[s] + al_d[d]) - dec_f32(emax[d]));
    float alpha = ex / (denom[d] + 1e-16f);
    if (lane < NCLS)
        atomicAdd(&agg[(size_t)d * NCLS + lane],
                  alpha * h2[(size_t)s * NCLS_PAD + lane]);
}

__global__ void out_kernel(const float* __restrict__ agg2,
                           const float* __restrict__ b2,
                           float* __restrict__ out) {
    int n = blockIdx.x * blockDim.x + threadIdx.x;
    if (n >= N_NODES) return;
    float v[NCLS];
    float m = -INFINITY;
    for (int c = 0; c < NCLS; ++c) {
        v[c] = agg2[(size_t)n * NCLS + c] + b2[c];
        m = fmaxf(m, v[c]);
    }
    float s = 0.0f;
    for (int c = 0; c < NCLS; ++c) s += expf(v[c] - m);
    float l = logf(s);
    for (int c = 0; c < NCLS; ++c) out[(size_t)n * NCLS + c] = v[c] - m - l;
}

// ---------- launcher ----------

extern "C" void kernel_launch(void* const* d_in, const int* in_sizes, int n_in,
                              void* d_out, int out_size, void* d_ws, size_t ws_size,
                              hipStream_t stream) {
    const float* x     = (const float*)d_in[0];
    const int*   ei    = (const int*)  d_in[1];
    const float* W1    = (const float*)d_in[2];
    const float* a1s   = (const float*)d_in[3];
    const float* a1d   = (const float*)d_in[4];
    const float* b1    = (const float*)d_in[5];
    const float* W2    = (const float*)d_in[6];
    const float* a2s   = (const float*)d_in[7];
    const float* a2d   = (const float*)d_in[8];
    const float* b2    = (const float*)d_in[9];
    float* out = (float*)d_out;

    // workspace partition (floats); atomic targets laid out contiguously for memset
    float* ws = (float*)d_ws;
    size_t o = 0;
    float*    h1     = ws + o; o += (size_t)N_NODES * D_FEAT;
    float*    al1_s  = ws + o; o += (size_t)N_NODES * HEADS;
    float*    al1_d  = ws + o; o += (size_t)N_NODES * HEADS;
    unsigned* emax1  = (unsigned*)(ws + o); o += (size_t)N_NODES * HEADS;
    float*    denom1 = ws + o; o += (size_t)N_NODES * HEADS;
    float*    agg1   = ws + o; o += (size_t)N_NODES * D_FEAT;     // reused as h2in
    float*    h2     = ws + o; o += (size_t)N_NODES * NCLS_PAD;
    float*    al2_s  = ws + o; o += (size_t)N_NODES;
    float*    al2_d  = ws + o; o += (size_t)N_NODES;
    unsigned* emax2  = (unsigned*)(ws + o); o += (size_t)N_NODES;
    float*    denom2 = ws + o; o += (size_t)N_NODES;
    float*    agg2   = ws + o; o += (size_t)N_NODES * NCLS;
    float*    W2p    = ws + o; o += (size_t)D_FEAT * NCLS_PAD;

    const int B = 256;

    // zero the atomic-accumulated regions (emax1|denom1|agg1 and emax2|denom2|agg2 are contiguous)
    (void)hipMemsetAsync(emax1, 0,
                   (size_t)(N_NODES * HEADS * 2 + N_NODES * D_FEAT) * sizeof(float), stream);
    (void)hipMemsetAsync(emax2, 0,
                   (size_t)(N_NODES * 2 + N_NODES * NCLS) * sizeof(float), stream);

    // padded W2 [128 x 32]
    build_w2p_kernel<<<(D_FEAT * NCLS_PAD + B - 1) / B, B, 0, stream>>>(W2, W2p);

    // layer 1: h1 = x @ W1  (3125 row-strips, 8 col tiles, fp32 WMMA, B staged in LDS)
    {
        int waves  = N_NODES / 16;
        int blocks = (waves + (B / 32) - 1) / (B / 32);
        size_t shbytes = (size_t)D_FEAT * (8 * 16 + 8) * sizeof(float);   // 128 x 136
        wmma_gemm_f32<8><<<blocks, B, shbytes, stream>>>(x, D_FEAT, W1,
                                                         h1, D_FEAT, N_NODES, D_FEAT);
    }
    al1_kernel<<<(N_NODES * HEADS + B - 1) / B, B, 0, stream>>>(h1, a1s, a1d, al1_s, al1_d);

    e1_max_kernel<<<(E_TOT + B - 1) / B, B, 0, stream>>>(ei, al1_s, al1_d, emax1);
    e1_sum_kernel<<<(E_TOT + B - 1) / B, B, 0, stream>>>(ei, al1_s, al1_d, emax1, denom1);
    {
        long long th = (long long)E_TOT * 32;
        e1_agg_kernel<<<(unsigned)((th + B - 1) / B), B, 0, stream>>>(
            ei, al1_s, al1_d, emax1, denom1, h1, agg1);
    }
    h2in_kernel<<<(N_NODES * D_FEAT + B - 1) / B, B, 0, stream>>>(agg1, b1);

    // layer 2: h2 = elu(agg1) @ W2p  (2 col tiles, fp32 WMMA, B staged in LDS)
    {
        int waves  = N_NODES / 16;
        int blocks = (waves + (B / 32) - 1) / (B / 32);
        size_t shbytes = (size_t)D_FEAT * (2 * 16 + 8) * sizeof(float);   // 128 x 40
        wmma_gemm_f32<2><<<blocks, B, shbytes, stream>>>(agg1, D_FEAT, W2p,
                                                         h2, NCLS_PAD, N_NODES, D_FEAT);
    }
    al2_kernel<<<(N_NODES + B - 1) / B, B, 0, stream>>>(h2, a2s, a2d, al2_s, al2_d);

    e2_max_kernel<<<(E_TOT + B - 1) / B, B, 0, stream>>>(ei, al2_s, al2_d, emax2);
    e2_sum_kernel<<<(E_TOT + B - 1) / B, B, 0, stream>>>(ei, al2_s, al2_d, emax2, denom2);
    {
        long long th = (long long)E_TOT * 32;
        e2_agg_kernel<<<(unsigned)((th + B - 1) / B), B, 0, stream>>>(
            ei, al2_s, al2_d, emax2, denom2, h2, agg2);
    }

    out_kernel<<<(N_NODES + B - 1) / B, B, 0, stream>>>(agg2, b2, out);
    (void)in_sizes; (void)n_in; (void)out_size; (void)ws_size;
}